// NumpyTransducerLoss_78391743087292
// MI455X (gfx1250) — compile-verified
//
#include <hip/hip_runtime.h>
#include <stdint.h>

namespace {

constexpr int kT  = 512;   // logit time steps
constexpr int kU  = 97;    // targets + 1
constexpr int kD  = 512;   // vocab
constexpr int kPF = 8;     // diagonals in flight (async prefetch depth)
constexpr int kRD = 16;    // LDS ring slots (power of two, > kPF)
constexpr uint64_t kStepBytes = (uint64_t)kU * kD * 4u;  // +1 diagonal => +1 t row

__device__ __forceinline__ float log_add_exp(float a, float b) {
  float hi = fmaxf(a, b);
  float lo = fminf(a, b);
  // exp underflows to 0 for large gaps and __logf(1.0f) == 0, so no branch.
  return hi + __logf(1.0f + __expf(lo - hi));
}

// One wave32 per batch element. Anti-diagonal wavefront DP over the [T,U]
// lattice: 607 serial logaddexp levels is the algorithmic floor. Operand
// gathers (blank + target log-prob per cell) run kPF diagonals ahead through
// CDNA5 async global->LDS loads (ASYNCcnt), keeping the ~60 MB of line-granular
// HBM traffic off the serial chain. Address streams advance by a constant
// per diagonal, so the steady-state loop is adds/compares/selects only.
__global__ void __launch_bounds__(32)
rnnt_loss_dp(const float* __restrict__ lp,      // [B,T,U,D] log-probs
             const int*  __restrict__ targets,  // [B,U-1]
             const int*  __restrict__ blank_p,  // [1]
             float* __restrict__ out)           // [B]
{
  // ring[slot][u] = {lb, le} operand pair of one lattice cell (8 B per lane,
  // lanes stride 2 dwords -> all 64 LDS banks hit exactly once per b64 read).
  __shared__ float ring[kRD][128][2];

  const int b    = blockIdx.x;
  const int lane = threadIdx.x;

  const int blank    = *blank_p;
  const int blankIdx = ((blank % kD) + kD) % kD;      // -1 -> D-1

  const float*   lpb  = lp + (size_t)b * kT * kU * kD;
  const uint64_t base = (uint64_t)(uintptr_t)lpb;
  const uint32_t ringBase = (uint32_t)(uintptr_t)(&ring[0][0][0]);

  // Per-(group,lane) column state: validity window over diagonals and the two
  // incrementally-advancing gather address streams.
  //   lb operand of cell (t,u): lp[t-1, u, blankIdx],  t = e - u
  //   le operand of cell (t,u): lp[t,   u-1, tgt[u-1]]
  // Cell (t,u) is live iff 1 <= t <= T-1 and u <= U-1  <=>
  //   (uint)(e - (u+1)) < T-1  (and span forced to 0 for u > U-1).
  uint64_t gaLB[4], gaLE[4];
  int      eLo[4];
  uint32_t span[4], u8[4];
#pragma unroll
  for (int g = 0; g < 4; ++g) {
    const int u = g * 32 + lane;
    u8[g]   = (uint32_t)(u * 8);
    eLo[g]  = u + 1;
    span[g] = (u <= kU - 1) ? (uint32_t)(kT - 1) : 0u;
    const int j  = u - 1;
    const int tg = (j >= 0 && j < kU - 1) ? targets[b * (kU - 1) + j] : 0;
    // streams initialized at diagonal e = 1 (t = 1 - u); wrapped arithmetic is
    // fine: masked lanes select `base` until their window opens.
    const int64_t idxLB = ((int64_t)(1 - u) - 1) * (kU * kD) + (int64_t)u * kD + blankIdx;
    const int64_t idxLE = ((int64_t)(1 - u)) * (kU * kD) + (int64_t)(u - 1) * kD + tg;
    gaLB[g] = base + (uint64_t)(idxLB * 4);
    gaLE[g] = base + (uint64_t)(idxLE * 4) - 4u;     // consumed with offset:4
  }

  int eIssue = 1;
  auto issue_diag = [&]() {
    const uint32_t sBase = ringBase + (uint32_t)((eIssue & (kRD - 1)) << 10);
#pragma unroll
    for (int g = 0; g < 4; ++g) {
      const bool v   = (uint32_t)(eIssue - eLo[g]) < span[g];
      const uint64_t alb = v ? gaLB[g] : base;       // clamped lanes hit lpb[0]
      const uint64_t ale = v ? gaLE[g] : base - 4u;  // +offset:4 -> lpb[0]
      const uint32_t la  = sBase + u8[g];
      asm volatile("global_load_async_to_lds_b32 %0, %1, off"
                   :: "v"(la), "v"(alb) : "memory");
      asm volatile("global_load_async_to_lds_b32 %0, %1, off offset:4"
                   :: "v"(la), "v"(ale) : "memory");
      gaLB[g] += kStepBytes;
      gaLE[g] += kStepBytes;
    }
    ++eIssue;
  };

  // ---- init row t=0 : alpha0[u] = sum_{k<u} le[0,k]  (exclusive wave scan)
  float A[4];   // A[g] = latest alpha of column u = g*32+lane
  {
    float e0[4];
#pragma unroll
    for (int g = 0; g < 4; ++g) {
      const int k = g * 32 + lane;
      if (k < kU - 1) {
        const int tg = targets[b * (kU - 1) + k];
        e0[g] = lpb[(size_t)k * kD + tg];
      } else {
        e0[g] = 0.0f;
      }
    }
    float carry = 0.0f;
#pragma unroll
    for (int g = 0; g < 4; ++g) {
      float v = e0[g];
#pragma unroll
      for (int off = 1; off < 32; off <<= 1) {
        float n = __shfl_up(v, off);
        if (lane >= off) v += n;
      }
      A[g]  = carry + v - e0[g];     // exclusive prefix across flattened u
      carry += __shfl(v, 31);        // group total
    }
  }

  // ---- warm the async pipeline: diagonals 1..kPF in flight (64 ops; the HW
  // throttles at the 6-bit ASYNCcnt limit, accounting stays exact).
  for (int i = 0; i < kPF; ++i) issue_diag();

  // ---- wavefront DP over anti-diagonals d = t+u, d = 1..T+U-2
  for (int d = 1; d <= kT + kU - 2; ++d) {
    // <= 8*(kPF-1) outstanding => all 8 gathers of diagonal d have landed.
    asm volatile("s_wait_asynccnt 56" ::: "memory");
    const char* rSlot = (const char*)ring + (uint32_t)((d & (kRD - 1)) << 10);

    float lbv[4], lev[4], rot[4];
#pragma unroll
    for (int g = 0; g < 4; ++g) {
      const float2 pr = *(const float2*)(rSlot + u8[g]);   // ds_load_b64
      lbv[g] = pr.x;
      lev[g] = pr.y;
    }
    // neighbor alpha[t,u-1] lives one lane left (previous group at lane 0)
#pragma unroll
    for (int g = 0; g < 4; ++g) rot[g] = __shfl(A[g], (lane + 31) & 31);

    float prev = 0.0f;
#pragma unroll
    for (int g = 0; g < 4; ++g) {
      const bool act = (uint32_t)(d - eLo[g]) < span[g];
      const float nb   = (lane == 0) ? prev : rot[g];
      const float skip = A[g] + lbv[g];
      float val = log_add_exp(skip, nb + lev[g]);
      if (g == 0) val = (lane == 0) ? skip : val;    // u == 0: blank-only entry
      A[g] = act ? val : A[g];
      prev = rot[g];
    }

    issue_diag();   // diagonal d+kPF; past the end every lane masks to lpb[0],
                    // keeping the 8-ops-per-diagonal ASYNCcnt invariant exact
  }

  if (lane == 0) {  // lane owning u = 96 (group 3)
    const float lb_last = lpb[((size_t)(kT - 1) * kU + (kU - 1)) * kD + blankIdx];
    out[b] = -(A[3] + lb_last);
  }
}

} // namespace

extern "C" void kernel_launch(void* const* d_in, const int* in_sizes, int n_in,
                              void* d_out, int out_size, void* d_ws, size_t ws_size,
                              hipStream_t stream) {
  (void)n_in; (void)d_ws; (void)ws_size; (void)out_size;
  const float* lp      = (const float*)d_in[0];   // log_probs [B,T,U,D] f32
  const int*   targets = (const int*)d_in[3];     // targets   [B,U-1]   i32
  const int*   blank_p = (const int*)d_in[4];     // blank scalar (== -1)
  float*       out     = (float*)d_out;           // costs [B] f32

  const int B = in_sizes[1];                      // logit_lengths has B entries
  rnnt_loss_dp<<<dim3(B), dim3(32), 0, stream>>>(lp, targets, blank_p, out);
}